// NeighborhoodSuperpixelAttention_16398185136401
// MI455X (gfx1250) — compile-verified
//
#include <hip/hip_runtime.h>
#include <hip/hip_bf16.h>

typedef __attribute__((ext_vector_type(2))) float v2f;
typedef __attribute__((ext_vector_type(8))) float v8f;

// Problem constants
constexpr int CB  = 2;            // batch
constexpr int CC  = 128;          // channels
constexpr int CH  = 96;
constexpr int CW  = 96;
constexpr int CNH = 4;            // heads
constexpr int CHD = 32;           // head dim
constexpr int CHW = CH * CW;      // 9216
constexpr int NPIX = CB * CHW;    // 18432

// ---------------------------------------------------------------------------
// Kernel 1: q/k/v = xs @ w   (xs read on the fly from NCHW x)
// One wave -> 16-pixel x 128-channel output tile of one of {q,k,v}.
// fp32 WMMA 16x16x4: A = xs tile (16 pix x 4 ch), B = weight (4 x 16).
// ---------------------------------------------------------------------------
__global__ __launch_bounds__(128) void qkv_gemm_kernel(
    const float* __restrict__ x,
    const float* __restrict__ wq, const float* __restrict__ wk,
    const float* __restrict__ wv,
    float* __restrict__ q, float* __restrict__ k, float* __restrict__ v)
{
    const int wave  = threadIdx.x >> 5;
    const int lane  = threadIdx.x & 31;
    const int l16   = lane & 15;
    const int khalf = lane >> 4;

    const int mtile = blockIdx.x * 4 + wave;      // 0..1151
    const int which = blockIdx.y;                 // 0:q 1:k 2:v
    const float* w  = (which == 0) ? wq : (which == 1) ? wk : wv;
    float*      out = (which == 0) ? q  : (which == 1) ? k  : v;

    const int rowbase = mtile * 16;               // pixel row base (16 | 9216)
    const int b = rowbase / CHW;
    const int p = rowbase - b * CHW;

    v8f acc[8] = {};                              // 16x128 tile accumulators
    const float* xbase = x + (size_t)b * CC * CHW + p + l16;

    for (int k0 = 0; k0 < CC; k0 += 4) {
        const int kk = k0 + khalf * 2;
        v2f a;                                    // A[m=l16][kk], A[m][kk+1]
        a.x = xbase[(size_t)kk * CHW];
        a.y = xbase[(size_t)(kk + 1) * CHW];
#pragma unroll
        for (int j = 0; j < 8; ++j) {
            v2f bf;                               // B[kk][n=j*16+l16]
            bf.x = w[kk * CC + j * 16 + l16];
            bf.y = w[(kk + 1) * CC + j * 16 + l16];
            acc[j] = __builtin_amdgcn_wmma_f32_16x16x4_f32(
                false, a, false, bf, (short)0, acc[j], false, false);
        }
    }

    // D layout: VGPR i -> M = i + 8*khalf ; N = l16
    float* obase = out + (size_t)rowbase * CC;
#pragma unroll
    for (int j = 0; j < 8; ++j) {
#pragma unroll
        for (int i = 0; i < 8; ++i) {
            const int mm = i + 8 * khalf;
            const int nn = j * 16 + l16;
            obase[(size_t)mm * CC + nn] = acc[j][i];
        }
    }
}

// ---------------------------------------------------------------------------
// Kernel 2: masked 7x7 neighborhood attention, LDS-tiled.
// Workgroup = 8x8 pixel tile x 1 head. Halo 14x14 x 32 dims of k and v
// (49 KB) staged in LDS -> 16x less global/L2 traffic than direct reads.
// Wave w owns tile row w; lane d owns head-dim d; online softmax.
// ---------------------------------------------------------------------------
__global__ __launch_bounds__(256) void nb_attn_lds_kernel(
    const float* __restrict__ q, const float* __restrict__ k,
    const float* __restrict__ v, const int* __restrict__ labels,
    float* __restrict__ out)
{
    __shared__ float sK[196 * 32];
    __shared__ float sV[196 * 32];
    __shared__ int   sL[196];

    const int tid  = threadIdx.x;
    const int wave = tid >> 5;
    const int lane = tid & 31;

    const int tile = blockIdx.x;                  // 0..143
    const int ty = tile / 12, tx = tile - ty * 12;
    const int r0 = ty * 8, c0 = tx * 8;
    const int b    = blockIdx.y >> 2;
    const int head = blockIdx.y & 3;

    int hr0 = r0 - 3; hr0 = hr0 < 0 ? 0 : (hr0 > CH - 14 ? CH - 14 : hr0);
    int hc0 = c0 - 3; hc0 = hc0 < 0 ? 0 : (hc0 > CW - 14 ? CW - 14 : hc0);

    const int hbase = b * CHW;
    const int coff  = head * CHD;

    // Cooperative halo load: 196 positions x 32 dims; 128B coalesced per wave.
    for (int i = tid; i < 196 * 32; i += 256) {
        const int pos = i >> 5;
        const int d   = i & 31;
        const int pr  = pos / 14;
        const int gr  = hr0 + pr;
        const int gc  = hc0 + pos - pr * 14;
        const size_t g = (size_t)(hbase + gr * CW + gc) * CC + coff + d;
        sK[i] = k[g];
        sV[i] = v[g];
    }
    for (int i = tid; i < 196; i += 256) {
        const int pr = i / 14;
        sL[i] = labels[hbase + (hr0 + pr) * CW + hc0 + i - pr * 14];
    }
    __syncthreads();

    const float scale = 0.17677669529663687f;     // 32^-0.5
    const int r = r0 + wave;                      // wave owns one tile row
    for (int pc = 0; pc < 8; ++pc) {
        const int c   = c0 + pc;
        const int pix = hbase + r * CW + c;
        int sr = r - 3; sr = sr < 0 ? 0 : (sr > CH - 7 ? CH - 7 : sr);
        int sc = c - 3; sc = sc < 0 ? 0 : (sc > CW - 7 ? CW - 7 : sc);
        const int lr0 = sr - hr0, lc0 = sc - hc0;

        const float qd    = q[(size_t)pix * CC + coff + lane] * scale;
        const int   mylab = sL[(r - hr0) * 14 + (c - hc0)];

        float mmax = -__builtin_huge_valf();
        float ssum = 0.0f, acc = 0.0f;
        for (int rr = 0; rr < 7; ++rr) {
            const int prow = (lr0 + rr) * 14 + lc0;
            for (int cc = 0; cc < 7; ++cc) {
                const int pos = prow + cc;
                const float kd = sK[pos * 32 + lane];
                float logit = qd * kd;
#pragma unroll
                for (int off = 16; off > 0; off >>= 1)
                    logit += __shfl_xor(logit, off, 32);
                if (sL[pos] != mylab) logit = -1000000000.0f;

                const float nm   = fmaxf(mmax, logit);
                const float corr = __expf(mmax - nm);
                const float e    = __expf(logit - nm);
                const float vd   = sV[pos * 32 + lane];
                ssum = ssum * corr + e;
                acc  = acc * corr + e * vd;
                mmax = nm;
            }
        }
        out[(size_t)pix * CC + coff + lane] = acc / ssum;
    }
}

// ---------------------------------------------------------------------------
// Kernel 3: final = (attn @ wproj) transposed to NCHW, computed as
//   D[c][pix] = sum_k wproj[k][c] * attn[pix][k]   (A = wproj^T, B = attn^T)
// so both weight loads and NCHW stores are lane-coalesced.
// ---------------------------------------------------------------------------
__global__ __launch_bounds__(128) void proj_gemm_kernel(
    const float* __restrict__ attn, const float* __restrict__ wp,
    float* __restrict__ out)
{
    const int wave  = threadIdx.x >> 5;
    const int lane  = threadIdx.x & 31;
    const int l16   = lane & 15;
    const int khalf = lane >> 4;

    const int pixbase = blockIdx.x * 16;                // 1152 tiles
    const int ctile   = blockIdx.y * 4 + wave;          // 0..7
    const int c0      = ctile * 16;
    const int b = pixbase / CHW;
    const int p = pixbase - b * CHW;

    v8f acc = {};
    for (int k0 = 0; k0 < CC; k0 += 4) {
        const int kk = k0 + khalf * 2;
        v2f a;                                          // A[m=c][k] = wp[k][c]
        a.x = wp[kk * CC + c0 + l16];
        a.y = wp[(kk + 1) * CC + c0 + l16];
        v2f bf;                                         // B[k][n=pix]
        bf.x = attn[(size_t)(pixbase + l16) * CC + kk];
        bf.y = attn[(size_t)(pixbase + l16) * CC + kk + 1];
        acc = __builtin_amdgcn_wmma_f32_16x16x4_f32(
            false, a, false, bf, (short)0, acc, false, false);
    }

    // D: VGPR i -> channel c0 + i + 8*khalf ; N = pixel offset l16
    float* ob = out + (size_t)b * CC * CHW + p + l16;
#pragma unroll
    for (int i = 0; i < 8; ++i) {
        const int c = c0 + i + 8 * khalf;
        ob[(size_t)c * CHW] = acc[i];
    }
}

// ---------------------------------------------------------------------------
extern "C" void kernel_launch(void* const* d_in, const int* in_sizes, int n_in,
                              void* d_out, int out_size, void* d_ws, size_t ws_size,
                              hipStream_t stream)
{
    (void)in_sizes; (void)n_in; (void)out_size; (void)ws_size;
    const float* x      = (const float*)d_in[0];
    const int*   labels = (const int*)d_in[1];
    const float* wq     = (const float*)d_in[2];
    const float* wk     = (const float*)d_in[3];
    const float* wv     = (const float*)d_in[4];
    const float* wp     = (const float*)d_in[5];

    float* q    = (float*)d_ws;
    float* k    = q + (size_t)NPIX * CC;
    float* v    = k + (size_t)NPIX * CC;
    float* attn = v + (size_t)NPIX * CC;
    float* out  = (float*)d_out;

    dim3 g1(NPIX / 16 / 4, 3);   // (288, 3), 4 waves/block
    qkv_gemm_kernel<<<g1, 128, 0, stream>>>(x, wq, wk, wv, q, k, v);

    dim3 g2((CH / 8) * (CW / 8), CB * CNH);   // (144, 8)
    nb_attn_lds_kernel<<<g2, 256, 0, stream>>>(q, k, v, labels, attn);

    dim3 g3(NPIX / 16, 2);       // (1152, 2), 4 waves/block
    proj_gemm_kernel<<<g3, 128, 0, stream>>>(attn, wp, out);
}